// FlashAttentionScore_50345606643921
// MI455X (gfx1250) — compile-verified
//
#include <hip/hip_runtime.h>
#include <math.h>

typedef __attribute__((ext_vector_type(2))) float v2f;
typedef __attribute__((ext_vector_type(8))) float v8f;
typedef __attribute__((ext_vector_type(4))) int   v4i;

#define S_LEN   2048
#define D_DIM   128
#define QB      128     // query rows per workgroup (8 waves x 16 rows)
#define KT      16      // key rows per LDS tile
#define KSTRIDE 132     // padded LDS row stride (floats) for K/V tiles: bank-conflict free
#define PSTRIDE 18      // padded LDS row stride (floats) for per-wave P tile
#define QK_SCALE 0.08838834764831845f   // 1/sqrt(128)

#define AS1 __attribute__((address_space(1)))
#define AS3 __attribute__((address_space(3)))

// Async global->LDS copy (16B), tracked by ASYNCcnt (gfx1250).
__device__ __forceinline__ void async_b128(const float* g, float* l) {
    __builtin_amdgcn_global_load_async_to_lds_b128(
        (AS1 v4i*)g, (AS3 v4i*)l, 0, 0);
}

// Each of the 256 threads streams 8 floats of K and 8 floats of V for one tile.
__device__ __forceinline__ void issue_tile_loads(const float* __restrict__ kh,
                                                 const float* __restrict__ vh,
                                                 float* __restrict__ ksbuf,
                                                 float* __restrict__ vsbuf,
                                                 int kt, int tid) {
    const int row = tid >> 4;            // 0..15
    const int col = (tid & 15) * 8;      // 0..120
    const float* kg = kh + (size_t)(kt * KT + row) * D_DIM + col;
    const float* vg = vh + (size_t)(kt * KT + row) * D_DIM + col;
    float* kl = ksbuf + row * KSTRIDE + col;
    float* vl = vsbuf + row * KSTRIDE + col;
    async_b128(kg,     kl);
    async_b128(kg + 4, kl + 4);
    async_b128(vg,     vl);
    async_b128(vg + 4, vl + 4);
}

__global__ __launch_bounds__(256, 1)
void fa_fwd_f32_wmma(const float* __restrict__ q,
                     const float* __restrict__ k,
                     const float* __restrict__ v,
                     float* __restrict__ out,
                     float* __restrict__ omax,
                     float* __restrict__ osum)
{
    __shared__ float Ks[2][16 * KSTRIDE];       // 2 x 8448 B (double buffered)
    __shared__ float Vs[2][16 * KSTRIDE];       // 2 x 8448 B
    __shared__ float Ps[8][16 * PSTRIDE];       // 9216 B (per-wave P tiles)

    const int nqb  = S_LEN / QB;                // 16 query blocks per head
    const int head = blockIdx.x / nqb;          // 0..31  (b*N + n)
    const int qb   = blockIdx.x % nqb;
    const int tid  = threadIdx.x;
    const int wave = tid >> 5;
    const int lane = tid & 31;
    const int half = lane >> 4;                 // 0: lanes 0-15, 1: lanes 16-31
    const int l16  = lane & 15;

    const size_t headoff = (size_t)head * S_LEN * D_DIM;
    const float* qh = q + headoff;
    const float* kh = k + headoff;
    const float* vh = v + headoff;

    const int qtile = qb * QB + wave * 16;      // first query row of this wave
    const int qrow  = qtile + l16;              // row held by this lane's A frags

    // ---- Load Q as 32 A-fragments (16x4 f32 each), covering D = 128 ----
    // A layout (32-bit 16x4): lane L holds row M=L%16; VGPR0/1 = K = 2*(L/16) + {0,1}
    v2f aq[32];
    {
        const float* qp = qh + (size_t)qrow * D_DIM + 2 * half;
        #pragma unroll
        for (int kk = 0; kk < 32; ++kk)
            aq[kk] = *(const v2f*)(qp + 4 * kk);
    }

    // ---- Flash-attention state ----
    v8f Oacc[8];                                // 16x128 f32 output accum (8 C tiles)
    #pragma unroll
    for (int d = 0; d < 8; ++d) Oacc[d] = (v8f){};
    float m_i[8], l_i[8];                       // per-row (C layout) running max / sum
    #pragma unroll
    for (int r = 0; r < 8; ++r) { m_i[r] = -__builtin_inff(); l_i[r] = 0.0f; }

    const int nkt = (qb + 1) * (QB / KT);       // key tiles covering [0, (qb+1)*128)

    // Prologue: prefetch tile 0 into buffer 0.
    issue_tile_loads(kh, vh, Ks[0], Vs[0], 0, tid);

    for (int kt = 0; kt < nkt; ++kt) {
        const int cur = kt & 1;

        // Tile kt's async loads (issued last iteration) must have landed,
        // for every wave, before any wave reads the buffer.
        asm volatile("s_wait_asynccnt 0x0" ::: "memory");
        __syncthreads();

        // Prefetch next tile into the other buffer; overlaps with compute below.
        // Safe: the barrier above means all waves finished reading buf cur^1
        // (its last readers were in iteration kt-1's compute).
        if (kt + 1 < nkt)
            issue_tile_loads(kh, vh, Ks[cur ^ 1], Vs[cur ^ 1], kt + 1, tid);

        // ---- S = Q @ K^T (16x16 tile), K-dim = 128 -> 32 f32 WMMAs ----
        // Two independent accumulator chains for matrix-pipe ILP.
        // B frag (4x16): lane holds col n=l16; rows K = 8kk + 2*half + {0,1} for b0, +4 for b1;
        // B[kd][n] = K[n][kd] => same addressing as A frags, read from Ks.
        v8f acc0 = (v8f){};
        v8f acc1 = (v8f){};
        {
            const float* kp = &Ks[cur][l16 * KSTRIDE + 2 * half];
            #pragma unroll
            for (int kk = 0; kk < 16; ++kk) {
                v2f b0 = *(const v2f*)(kp + 8 * kk);
                v2f b1 = *(const v2f*)(kp + 8 * kk + 4);
                acc0 = __builtin_amdgcn_wmma_f32_16x16x4_f32(
                         false, aq[2 * kk + 0], false, b0, (short)0, acc0, false, false);
                acc1 = __builtin_amdgcn_wmma_f32_16x16x4_f32(
                         false, aq[2 * kk + 1], false, b1, (short)0, acc1, false, false);
            }
        }
        v8f acc = acc0 + acc1;

        // ---- scale, causal mask, online softmax (per C-layout row) ----
        const int kbase = kt * KT;
        float alpha[8];
        #pragma unroll
        for (int r = 0; r < 8; ++r) {
            float s = acc[r] * QK_SCALE;
            const int myrow = qtile + r + 8 * half;
            const int kcol  = kbase + l16;
            if (kcol > myrow) s = -__builtin_inff();
            // row max over the 16 lanes of this half-wave
            float mx = s;
            mx = fmaxf(mx, __shfl_xor(mx, 1, 32));
            mx = fmaxf(mx, __shfl_xor(mx, 2, 32));
            mx = fmaxf(mx, __shfl_xor(mx, 4, 32));
            mx = fmaxf(mx, __shfl_xor(mx, 8, 32));
            const float mn = fmaxf(m_i[r], mx);
            const float al = __expf(m_i[r] - mn);
            const float pe = __expf(s - mn);     // exp(-inf)=0 for masked
            float rs = pe;
            rs += __shfl_xor(rs, 1, 32);
            rs += __shfl_xor(rs, 2, 32);
            rs += __shfl_xor(rs, 4, 32);
            rs += __shfl_xor(rs, 8, 32);
            l_i[r]   = l_i[r] * al + rs;
            m_i[r]   = mn;
            alpha[r] = al;
            // stash P in per-wave LDS tile: row M = r + 8*half, col N = l16
            Ps[wave][(r + 8 * half) * PSTRIDE + l16] = pe;
        }

        // rescale running output accumulators
        #pragma unroll
        for (int d = 0; d < 8; ++d)
            #pragma unroll
            for (int r = 0; r < 8; ++r)
                Oacc[d][r] *= alpha[r];

        // same-wave LDS RAW (cross-lane): make P visible before A-frag reads
        asm volatile("s_wait_dscnt 0x0" ::: "memory");

        // ---- O += P @ V : K-dim = 16 -> 4 A-frags, 8 independent depth tiles ----
        v2f ap[4];
        {
            const float* pp = &Ps[wave][l16 * PSTRIDE + 2 * half];
            #pragma unroll
            for (int kk = 0; kk < 4; ++kk)
                ap[kk] = *(const v2f*)(pp + 4 * kk);
        }
        #pragma unroll
        for (int d = 0; d < 8; ++d) {
            // B frag from V: B[kd][n] = V[kd][d*16 + n]; rows kd0, kd0+1 differ by KSTRIDE
            const float* vp = &Vs[cur][(2 * half) * KSTRIDE + d * 16 + l16];
            #pragma unroll
            for (int kk = 0; kk < 4; ++kk) {
                v2f b;
                b.x = vp[(4 * kk + 0) * KSTRIDE];
                b.y = vp[(4 * kk + 1) * KSTRIDE];
                Oacc[d] = __builtin_amdgcn_wmma_f32_16x16x4_f32(
                            false, ap[kk], false, b, (short)0, Oacc[d], false, false);
            }
        }
    }

    // ---- finalize: O /= l, write out + stats ----
    #pragma unroll
    for (int r = 0; r < 8; ++r) {
        const float inv = 1.0f / l_i[r];
        #pragma unroll
        for (int d = 0; d < 8; ++d) Oacc[d][r] *= inv;
    }

    float* oh = out + headoff;
    #pragma unroll
    for (int d = 0; d < 8; ++d) {
        #pragma unroll
        for (int r = 0; r < 8; ++r) {
            const int row = qtile + r + 8 * half;
            oh[(size_t)row * D_DIM + d * 16 + l16] = Oacc[d][r];
        }
    }

    // stats are replicated x8 along last dim; lanes l16<8 each write one copy
    if (l16 < 8) {
        const size_t sbase = (size_t)head * S_LEN * 8;
        #pragma unroll
        for (int r = 0; r < 8; ++r) {
            const int row = qtile + r + 8 * half;
            omax[sbase + (size_t)row * 8 + l16] = m_i[r];
            osum[sbase + (size_t)row * 8 + l16] = l_i[r];
        }
    }
}

extern "C" void kernel_launch(void* const* d_in, const int* in_sizes, int n_in,
                              void* d_out, int out_size, void* d_ws, size_t ws_size,
                              hipStream_t stream) {
    const float* q = (const float*)d_in[0];
    const float* k = (const float*)d_in[1];
    const float* v = (const float*)d_in[2];

    float* out  = (float*)d_out;
    float* omax = out  + (size_t)2 * 16 * 2048 * 128;   // B*N*S*D
    float* osum = omax + (size_t)2 * 16 * 2048 * 8;     // B*N*S*8

    const int heads = 2 * 16;            // B*N
    const int nqb   = S_LEN / QB;        // 16
    dim3 grid(heads * nqb);              // 512 workgroups
    dim3 block(256);                     // 8 waves (wave32)
    fa_fwd_f32_wmma<<<grid, block, 0, stream>>>(q, k, v, out, omax, osum);
}